// PUGCN_88948772700679
// MI455X (gfx1250) — compile-verified
//
#include <hip/hip_runtime.h>
#include <hip/hip_bf16.h>
#include <stdint.h>

// ---------------------------------------------------------------------------
// PU-GCN on MI455X (gfx1250): EdgeConv rewritten as dense WMMA-f32 GEMMs
// (P = X@(Wa-Wb)+b, Q = X@Wb) + gather/max epilogues.
// ---------------------------------------------------------------------------

#define NPTS   8192
#define CCH    128
#define KNB    20
#define KFULL  40      // K * max(DILS)
#define KNN_K  41      // KFULL + 1 (self)
#define RUP    4
#define RC     512     // R * C
#define M2     (NPTS * RUP)

typedef float v2f __attribute__((ext_vector_type(2)));
typedef float v8f __attribute__((ext_vector_type(8)));

// ---------------------------------------------------------------------------
// Weight preparation: split each EdgeConv weight [2C, Cout] into
// Wd = W[:C]-W[C:], Wb = W[C:] so EdgeConv becomes two dense GEMMs.
// ---------------------------------------------------------------------------
__global__ void prep_weights(const float* __restrict__ W0,
                             const float* __restrict__ Wg,
                             const float* __restrict__ Wu,
                             float* __restrict__ Wld, float* __restrict__ Wlb,
                             float* __restrict__ Wgd, float* __restrict__ Wgb,
                             float* __restrict__ Wud, float* __restrict__ Wub) {
  int stride = gridDim.x * blockDim.x;
  int g = blockIdx.x * blockDim.x + threadIdx.x;
  // lift: W0 [6, C] -> rows 0..2 minus rows 3..5
  for (int t = g; t < 3 * CCH; t += stride) {
    float a = W0[t], b = W0[3 * CCH + t];
    Wld[t] = a - b; Wlb[t] = b;
  }
  // 4 GCN layers: Wg [4, 2C, C]
  for (int t = g; t < 4 * CCH * CCH; t += stride) {
    int li = t >> 14, r = t & (CCH * CCH - 1);
    float a = Wg[li * 2 * CCH * CCH + r];
    float b = Wg[li * 2 * CCH * CCH + CCH * CCH + r];
    Wgd[t] = a - b; Wgb[t] = b;
  }
  // upsampler: Wu [2C, RC]
  for (int t = g; t < CCH * RC; t += stride) {
    float a = Wu[t], b = Wu[CCH * RC + t];
    Wud[t] = a - b; Wub[t] = b;
  }
}

// ---------------------------------------------------------------------------
// Squared norms of positions
// ---------------------------------------------------------------------------
__global__ void sqnorm_kernel(const float* __restrict__ pos, float* __restrict__ sq) {
  int i = blockIdx.x * blockDim.x + threadIdx.x;
  if (i < NPTS) {
    float x = pos[i * 3], y = pos[i * 3 + 1], z = pos[i * 3 + 2];
    sq[i] = x * x + y * y + z * z;
  }
}

// ---------------------------------------------------------------------------
// Brute-force kNN: one thread per query point; candidate tiles staged in LDS;
// per-thread sorted insertion list of 41 entries lives in LDS, laid out
// [slot][tid] so the 64 LDS banks are hit conflict-free. Strict-< insertion
// reproduces jax top_k's ascending-index tie break. Entry 0 (self) dropped.
// ---------------------------------------------------------------------------
__global__ void knn_kernel(const float* __restrict__ pos, const float* __restrict__ sq,
                           int* __restrict__ idxOut) {
  const int T = 128;
  __shared__ float cx[T], cy[T], cz[T], cs[T];
  __shared__ float dl[KNN_K * T];
  __shared__ int   il[KNN_K * T];
  int tid = threadIdx.x;
  int i = blockIdx.x * T + tid;
  float px = pos[i * 3], py = pos[i * 3 + 1], pz = pos[i * 3 + 2];
  float psq = sq[i];
  for (int s = 0; s < KNN_K; ++s) dl[s * T + tid] = 3.4e38f;
  for (int base = 0; base < NPTS; base += T) {
    __syncthreads();
    int j = base + tid;
    cx[tid] = pos[j * 3]; cy[tid] = pos[j * 3 + 1]; cz[tid] = pos[j * 3 + 2];
    cs[tid] = sq[j];
    __syncthreads();
    float worst = dl[(KNN_K - 1) * T + tid];
    for (int t = 0; t < T; ++t) {
      float d = psq + cs[t] - 2.0f * (px * cx[t] + py * cy[t] + pz * cz[t]);
      if (d < worst) {
        int p = KNN_K - 1;
        while (p > 0 && dl[(p - 1) * T + tid] > d) {
          dl[p * T + tid] = dl[(p - 1) * T + tid];
          il[p * T + tid] = il[(p - 1) * T + tid];
          --p;
        }
        dl[p * T + tid] = d;
        il[p * T + tid] = base + t;
        worst = dl[(KNN_K - 1) * T + tid];
      }
    }
  }
  for (int t = 0; t < KNN_K - 1; ++t)
    idxOut[(size_t)i * KFULL + t] = il[(t + 1) * T + tid];
}

// ---------------------------------------------------------------------------
// Generic fp32 WMMA GEMM:  Y[M,Nn] = act( X[M,Kd] @ W[Kd,Nn] + bias )
// act(v) = fmaxf(v, slope*v): slope=1 -> identity, slope in (0,1) -> LeakyReLU.
// One 16x16 output tile per wave (8 waves / 256-thread block), fp32 WMMA
// with K=4 per V_WMMA_F32_16X16X4_F32 step. Lane layout per CDNA5 ISA:
// lanes 0-15 carry K=k0..k0+1, lanes 16-31 carry K=k0+2..k0+3.
// ---------------------------------------------------------------------------
__global__ void wmma_gemm_f32(const float* __restrict__ X, const float* __restrict__ W,
                              const float* __restrict__ bias, float* __restrict__ Y,
                              int M, int Kd, int Nn, float slope) {
  int wave = threadIdx.x >> 5;
  int lane = threadIdx.x & 31;
  int tilesN = Nn >> 4;
  int totalTiles = (M >> 4) * tilesN;
  int tile = blockIdx.x * (blockDim.x >> 5) + wave;
  if (tile >= totalTiles) return;              // wave-uniform: EXEC stays all-1s
  int tm = tile / tilesN, tn = tile - tm * tilesN;
  int l15 = lane & 15, half = lane >> 4;

  const float* xrow = X + (size_t)(tm * 16 + l15) * Kd;
  const float* wcol = W + tn * 16 + l15;

  v8f c = {};
  for (int k0 = 0; k0 < Kd; k0 += 4) {
    int kb = k0 + 2 * half;
    v2f a; a.x = xrow[kb];                 a.y = xrow[kb + 1];
    v2f b; b.x = wcol[(size_t)kb * Nn];    b.y = wcol[(size_t)(kb + 1) * Nn];
    c = __builtin_amdgcn_wmma_f32_16x16x4_f32(false, a, false, b, (short)0, c,
                                              false, false);
  }

  int col = tn * 16 + l15;
  float bv = bias ? bias[col] : 0.0f;
#pragma unroll
  for (int r = 0; r < 8; ++r) {
    int row = tm * 16 + r + 8 * half;
    float v = c[r] + bv;
    v = fmaxf(v, v * slope);
    Y[(size_t)row * Nn + col] = v;
  }
}

// ---------------------------------------------------------------------------
// Lift EdgeConv P/Q (Kdim = 3, too small for WMMA): plain VALU.
// P0 = pos @ Wld + b0 ; Q0 = pos @ Wlb
// ---------------------------------------------------------------------------
__global__ void lift_pq(const float* __restrict__ pos,
                        const float* __restrict__ Wld, const float* __restrict__ Wlb,
                        const float* __restrict__ b0,
                        float* __restrict__ P0, float* __restrict__ Q0) {
  int stride = gridDim.x * blockDim.x;
  for (int t = blockIdx.x * blockDim.x + threadIdx.x; t < NPTS * CCH; t += stride) {
    int i = t / CCH, c = t - (t / CCH) * CCH;
    float x0 = pos[i * 3], x1 = pos[i * 3 + 1], x2 = pos[i * 3 + 2];
    P0[t] = x0 * Wld[c] + x1 * Wld[CCH + c] + x2 * Wld[2 * CCH + c] + b0[c];
    Q0[t] = x0 * Wlb[c] + x1 * Wlb[CCH + c] + x2 * Wlb[2 * CCH + c];
  }
}

// ---------------------------------------------------------------------------
// EdgeConv epilogue: out[i,c] (=|+=) max_j LReLU_0.2(P[i,c] + Q[nbr_j, c])
// One block per point, one thread per channel; neighbor indices staged in LDS.
// ---------------------------------------------------------------------------
__global__ void edge_max(const float* __restrict__ P, const float* __restrict__ Q,
                         const int* __restrict__ nbr, int dil,
                         float* __restrict__ out, int Cout, int accumulate) {
  __shared__ int nb[KNB];
  int i = blockIdx.x;
  int c = threadIdx.x;
  if (c < KNB) nb[c] = nbr[(size_t)i * KFULL + c * dil];
  __syncthreads();
  float p = P[(size_t)i * Cout + c];
  float m = -3.4e38f;
#pragma unroll 4
  for (int j = 0; j < KNB; ++j) {
    float v = p + Q[(size_t)nb[j] * Cout + c];
    v = fmaxf(v, 0.2f * v);
    m = fmaxf(m, v);
  }
  size_t o = (size_t)i * Cout + c;
  if (accumulate) out[o] += m; else out[o] = m;
}

// ---------------------------------------------------------------------------
// Residual combine: h += 0.5 * acc
// ---------------------------------------------------------------------------
__global__ void combine_residual(float* __restrict__ h, const float* __restrict__ acc,
                                 int total) {
  int stride = gridDim.x * blockDim.x;
  for (int t = blockIdx.x * blockDim.x + threadIdx.x; t < total; t += stride)
    h[t] += 0.5f * acc[t];
}

// ---------------------------------------------------------------------------
// Final reconstructor layer: out[M2,3] = q1[M2,128] @ Wr2[128,3] + br2
// ---------------------------------------------------------------------------
__global__ void recon_out(const float* __restrict__ q1, const float* __restrict__ Wr2,
                          const float* __restrict__ br2, float* __restrict__ out) {
  int r = blockIdx.x * blockDim.x + threadIdx.x;
  if (r >= M2) return;
  const float* row = q1 + (size_t)r * CCH;
  float s0 = br2[0], s1 = br2[1], s2 = br2[2];
#pragma unroll 8
  for (int c = 0; c < CCH; ++c) {
    float v = row[c];
    s0 += v * Wr2[c * 3];
    s1 += v * Wr2[c * 3 + 1];
    s2 += v * Wr2[c * 3 + 2];
  }
  out[r * 3]     = s0;
  out[r * 3 + 1] = s1;
  out[r * 3 + 2] = s2;
}

// ---------------------------------------------------------------------------
static inline int gemm_blocks(int M, int Nn) {
  return ((M >> 4) * (Nn >> 4) + 7) / 8;   // 8 waves (tiles) per 256-thread block
}

extern "C" void kernel_launch(void* const* d_in, const int* in_sizes, int n_in,
                              void* d_out, int out_size, void* d_ws, size_t ws_size,
                              hipStream_t stream) {
  const float* x   = (const float*)d_in[0];
  const float* W0  = (const float*)d_in[1];
  const float* b0  = (const float*)d_in[2];
  const float* Wg  = (const float*)d_in[3];
  const float* bg  = (const float*)d_in[4];
  const float* Wu  = (const float*)d_in[5];
  const float* bu  = (const float*)d_in[6];
  const float* Wr1 = (const float*)d_in[7];
  const float* br1 = (const float*)d_in[8];
  const float* Wr2 = (const float*)d_in[9];
  const float* br2 = (const float*)d_in[10];
  float* out = (float*)d_out;

  // ---- workspace carve (256B aligned) ----
  uintptr_t cur = (uintptr_t)d_ws;
  auto alloc = [&](size_t bytes) -> void* {
    cur = (cur + 255) & ~(uintptr_t)255;
    void* p = (void*)cur;
    cur += bytes;
    return p;
  };
  float* sq   = (float*)alloc(NPTS * sizeof(float));
  int*   idx  = (int*)  alloc((size_t)NPTS * KFULL * sizeof(int));
  float* h    = (float*)alloc((size_t)NPTS * CCH * sizeof(float));
  float* P    = (float*)alloc((size_t)NPTS * CCH * sizeof(float));
  float* Q    = (float*)alloc((size_t)NPTS * CCH * sizeof(float));
  float* acc  = (float*)alloc((size_t)NPTS * CCH * sizeof(float));
  float* Pu   = (float*)alloc((size_t)NPTS * RC * sizeof(float));
  float* Qu   = (float*)alloc((size_t)NPTS * RC * sizeof(float));
  float* u    = (float*)alloc((size_t)NPTS * RC * sizeof(float));   // == [M2, C]
  float* q1   = (float*)alloc((size_t)M2 * CCH * sizeof(float));
  float* Wld  = (float*)alloc(3 * CCH * sizeof(float));
  float* Wlb  = (float*)alloc(3 * CCH * sizeof(float));
  float* Wgd  = (float*)alloc((size_t)4 * CCH * CCH * sizeof(float));
  float* Wgb  = (float*)alloc((size_t)4 * CCH * CCH * sizeof(float));
  float* Wud  = (float*)alloc((size_t)CCH * RC * sizeof(float));
  float* Wub  = (float*)alloc((size_t)CCH * RC * sizeof(float));

  // ---- 0) derived weights ----
  prep_weights<<<256, 256, 0, stream>>>(W0, Wg, Wu, Wld, Wlb, Wgd, Wgb, Wud, Wub);

  // ---- 1) kNN graph ----
  sqnorm_kernel<<<NPTS / 256, 256, 0, stream>>>(x, sq);
  knn_kernel<<<NPTS / 128, 128, 0, stream>>>(x, sq, idx);

  // ---- 2) lift EdgeConv (6 -> C) ----
  lift_pq<<<512, 256, 0, stream>>>(x, Wld, Wlb, b0, P, Q);
  edge_max<<<NPTS, CCH, 0, stream>>>(P, Q, idx, /*dil=*/1, h, CCH, /*acc=*/0);

  // ---- 3) Inception DenseGCN blocks ----
  for (int blk = 0; blk < 2; ++blk) {
    hipMemsetAsync(acc, 0, (size_t)NPTS * CCH * sizeof(float), stream);
    for (int di = 0; di < 2; ++di) {
      int li  = blk * 2 + di;
      int dil = (di == 0) ? 1 : 2;
      wmma_gemm_f32<<<gemm_blocks(NPTS, CCH), 256, 0, stream>>>(
          h, Wgd + (size_t)li * CCH * CCH, bg + (size_t)li * CCH, P,
          NPTS, CCH, CCH, 1.0f);
      wmma_gemm_f32<<<gemm_blocks(NPTS, CCH), 256, 0, stream>>>(
          h, Wgb + (size_t)li * CCH * CCH, nullptr, Q,
          NPTS, CCH, CCH, 1.0f);
      edge_max<<<NPTS, CCH, 0, stream>>>(P, Q, idx, dil, acc, CCH, /*acc=*/1);
    }
    combine_residual<<<512, 256, 0, stream>>>(h, acc, NPTS * CCH);
  }

  // ---- 4) NodeShuffle upsampler EdgeConv (2C -> R*C) ----
  wmma_gemm_f32<<<gemm_blocks(NPTS, RC), 256, 0, stream>>>(
      h, Wud, bu, Pu, NPTS, CCH, RC, 1.0f);
  wmma_gemm_f32<<<gemm_blocks(NPTS, RC), 256, 0, stream>>>(
      h, Wub, nullptr, Qu, NPTS, CCH, RC, 1.0f);
  edge_max<<<NPTS, RC, 0, stream>>>(Pu, Qu, idx, /*dil=*/1, u, RC, /*acc=*/0);

  // ---- 5) reconstructor MLP: u is [M2, C] by row-major reshape ----
  wmma_gemm_f32<<<gemm_blocks(M2, CCH), 256, 0, stream>>>(
      u, Wr1, br1, q1, M2, CCH, CCH, 0.01f);
  recon_out<<<(M2 + 255) / 256, 256, 0, stream>>>(q1, Wr2, br2, out);
}